// MessagePassing_25357486916272
// MI455X (gfx1250) — compile-verified
//
#include <hip/hip_runtime.h>

// Problem sizes (fixed by the reference)
#define B_    64
#define L_    350
#define K_    8
#define D_    300
#define C_    20
#define PAD_  1
#define CHUNK_ 14   // L-chunk per block
#define NCH_   25   // 25 * 14 = 350
#define NPAD_ 32    // N padded to two 16-wide WMMA tiles

typedef __attribute__((ext_vector_type(2))) float v2f;
typedef __attribute__((ext_vector_type(8))) float v8f;

// ---------------------------------------------------------------------------
// Kernel 1: streaming masked-max message pass + gated combine, reduced over L.
// grid = B*NCH blocks, 320 threads (10 wave32). Thread t owns feature d = t.
// Memory-bound: coalesced f32 reads of embedded_neighbor_node dominate
// (~242 MB total -> ~10.4 us at 23.3 TB/s HBM).
// ---------------------------------------------------------------------------
__global__ __launch_bounds__(320) void mp_reduce(
    const int*   __restrict__ node_sets,   // [B,L]
    const float* __restrict__ e_node,      // [B,L,D]
    const float* __restrict__ e_w,         // [B,L,K]
    const float* __restrict__ e_nn,        // [B,L,K,D]
    const float* __restrict__ ir_tab,      // [V]
    float*       __restrict__ s)           // [B,D] accumulator (pre-zeroed)
{
    const int blk = blockIdx.x;
    const int b   = blk / NCH_;
    const int ch  = blk - b * NCH_;
    const int d   = threadIdx.x;
    const bool active = d < D_;

    float acc = 0.0f;
    const int l0 = ch * CHUNK_;

    for (int li = 0; li < CHUNK_; ++li) {
        const long base = (long)b * L_ + (l0 + li);

        // Uniform (per-block-iteration) scalars -> SMEM loads
        const int   node = node_sets[base];
        const float ir   = (node == PAD_) ? 1.0f : ir_tab[node];
        float w[K_];
        #pragma unroll
        for (int k = 0; k < K_; ++k) w[k] = e_w[base * K_ + k];

        if (active) {
            const float* np = e_nn + base * (long)(K_ * D_) + d;
            // prefetch next node's neighbor block (speculative, cacheline grain)
            if (li + 1 < CHUNK_)
                __builtin_prefetch(np + (long)(K_ * D_), 0, 1);

            float m = -1e18f;
            #pragma unroll
            for (int k = 0; k < K_; ++k) {
                float t = w[k] * np[(long)k * D_];
                t = (t == 0.0f) ? -1e18f : t;   // masked_fill(tmp==0, -1e18)
                m = fmaxf(m, t);
            }
            const float en = e_node[base * D_ + d];
            acc += (1.0f - ir) * m + ir * en;
        }
    }
    if (active) atomicAdd(&s[b * D_ + d], acc);   // global_atomic_add_f32
}

// ---------------------------------------------------------------------------
// Kernel 2: out[64,20] = softmax(relu(s @ W^T + b)) via V_WMMA_F32_16X16X4_F32.
// One block, 8 wave32; wave w computes tile (mtile = w>>1, ntile = w&1).
// Both operands are staged in LDS first (W zero-padded to 32 rows at store
// time), so the WMMA k-loop is pure ds_load_b64 with uniform control flow --
// no divergent EXEC-masked global loads in the hot loop.
// A: 16x4 f32  (lane<16: M=lane, v0=K0,v1=K1; lane>=16: M=lane-16, v0=K2,v1=K3)
// B: 4x16 f32  (v0: K0 / K2 halves, v1: K1 / K3), C/D: 8 VGPRs per ISA layout.
// ---------------------------------------------------------------------------
__global__ __launch_bounds__(256) void mp_gemm_softmax(
    const float* __restrict__ s,     // [64,300]
    const float* __restrict__ Wm,    // [20,300]
    const float* __restrict__ bias,  // [20]
    float*       __restrict__ out)   // [64,20]
{
    __shared__ float sA[B_ * D_];        // 76800 B
    __shared__ float sB[NPAD_ * D_];     // 38400 B (rows 20..31 zero)
    __shared__ float tile[B_][C_];       //  5120 B

    const int tid = threadIdx.x;

    // Cooperative, coalesced staging (uniform control flow, no divergence)
    for (int i = tid; i < B_ * D_; i += 256) sA[i] = s[i];
    for (int i = tid; i < NPAD_ * D_; i += 256)
        sB[i] = (i < C_ * D_) ? Wm[i] : 0.0f;   // zero-pad rows 20..31 here
    __syncthreads();

    const int lane  = tid & 31;
    const int wave  = tid >> 5;      // 0..7
    const int mtile = wave >> 1;     // 0..3
    const int ntile = wave & 1;      // 0..1
    const int hi    = lane >> 4;     // 0 or 1 (K-half selector)
    const int lo    = lane & 15;

    const int arow = mtile * 16 + lo;    // 0..63
    const int bcol = ntile * 16 + lo;    // 0..31 (padded rows read zeros)

    v8f c = {};
    for (int k0 = 0; k0 < D_; k0 += 4) {     // 75 exact steps, D=300
        const int ka = k0 + 2 * hi;
        v2f a, bm;
        a.x  = sA[arow * D_ + ka];
        a.y  = sA[arow * D_ + ka + 1];
        bm.x = sB[bcol * D_ + ka];
        bm.y = sB[bcol * D_ + ka + 1];
        // (neg_a, A, neg_b, B, c_mod, C, reuse_a, reuse_b)
        c = __builtin_amdgcn_wmma_f32_16x16x4_f32(
                false, a, false, bm, (short)0, c, false, false);
    }

    // D-tile layout: VGPR j -> M = j (lanes 0-15) / j+8 (lanes 16-31), N = lo
    #pragma unroll
    for (int j = 0; j < 8; ++j) {
        const int m = mtile * 16 + j + 8 * hi;
        if (bcol < C_) tile[m][bcol] = c[j];
    }
    __syncthreads();

    // Per-row bias + ReLU + softmax (64 rows, 20 cols)
    if (tid < B_) {
        float v[C_];
        float mx = -1e30f;
        #pragma unroll
        for (int n = 0; n < C_; ++n) {
            float x = tile[tid][n] + bias[n];
            x = fmaxf(x, 0.0f);
            v[n] = x;
            mx = fmaxf(mx, x);
        }
        float sum = 0.0f;
        #pragma unroll
        for (int n = 0; n < C_; ++n) { v[n] = __expf(v[n] - mx); sum += v[n]; }
        const float inv = 1.0f / sum;
        #pragma unroll
        for (int n = 0; n < C_; ++n) out[tid * C_ + n] = v[n] * inv;
    }
}

// ---------------------------------------------------------------------------
extern "C" void kernel_launch(void* const* d_in, const int* in_sizes, int n_in,
                              void* d_out, int out_size, void* d_ws, size_t ws_size,
                              hipStream_t stream) {
    const int*   node_sets = (const int*)  d_in[0];  // [B,L] int32
    const float* e_node    = (const float*)d_in[1];  // [B,L,D]
    const float* e_w       = (const float*)d_in[2];  // [B,L,K]
    const float* e_nn      = (const float*)d_in[3];  // [B,L,K,D]
    const float* ir_tab    = (const float*)d_in[4];  // [V,1]
    const float* Wm        = (const float*)d_in[5];  // [C,D]
    const float* bias      = (const float*)d_in[6];  // [C]
    float* out = (float*)d_out;                      // [B,C]
    float* s   = (float*)d_ws;                       // [B,D] scratch

    hipMemsetAsync(s, 0, (size_t)(B_ * D_) * sizeof(float), stream);
    mp_reduce<<<B_ * NCH_, 320, 0, stream>>>(node_sets, e_node, e_w, e_nn, ir_tab, s);
    mp_gemm_softmax<<<1, 256, 0, stream>>>(s, Wm, bias, out);
}